// E86InputMatrixDeltaCell_81819126989234
// MI455X (gfx1250) — compile-verified
//
#include <hip/hip_runtime.h>
#include <hip/hip_bf16.h>

// Problem constants (from reference): T=1024, B=8, H=8, N=32
constexpr int T_ = 1024;
constexpr int B_ = 8;
constexpr int H_ = 8;
constexpr int N_ = 32;
constexpr int D_ = H_ * N_ * N_;         // 8192 floats per (t,b)
constexpr int TILE_F = N_ * N_;          // 1024 floats per (b,h) tile
constexpr int ROWP = 36;                 // padded LDS row stride (144B, 16B aligned, conflict-light)
constexpr int TILE_P = N_ * ROWP;        // 1152 floats per LDS tile slot
constexpr int NSTAGES = 8;               // async prefetch depth (8 tiles * 8 B128 ops = 64 in flight max)

#define AS1 __attribute__((address_space(1)))
#define AS3 __attribute__((address_space(3)))

typedef int v4i __attribute__((ext_vector_type(4)));

// ---- CDNA5 async global->LDS copy (ASYNCcnt path) ----------------------------
__device__ __forceinline__ void async_copy_b128(const float* g, AS3 float* l) {
#if __has_builtin(__builtin_amdgcn_global_load_async_to_lds_b128)
  // signature (probe-confirmed via diagnostic): (v4i AS1*, v4i AS3*, imm offset, imm cpol)
  __builtin_amdgcn_global_load_async_to_lds_b128((AS1 v4i*)g, (AS3 v4i*)l,
                                                 /*offset=*/0, /*cpol=*/0);
#else
  // GV mode: vdst = LDS byte address VGPR, vaddr = 64-bit global address pair
  asm volatile("global_load_async_to_lds_b128 %0, %1, off"
               :: "v"(l), "v"((const AS1 float*)g)
               : "memory");
#endif
}

// Wait until at most `n` async ops outstanding. asm form guarantees the
// compiler cannot hoist subsequent LDS reads above the wait.
#define WAIT_ASYNCCNT_LE(n) asm volatile("s_wait_asynccnt %0" ::"n"(n) : "memory")

// ---- gfx1250 hardware tanh (transcendental unit) -----------------------------
__device__ __forceinline__ float fast_tanh(float x) {
#if __has_builtin(__builtin_amdgcn_tanhf)
  return __builtin_amdgcn_tanhf(x);
#elif __has_builtin(__builtin_amdgcn_tanh_f32)
  return __builtin_amdgcn_tanh_f32(x);
#else
  return tanhf(x);
#endif
}

__device__ __forceinline__ float fast_sigmoid(float x) {
  return 1.0f / (1.0f + __expf(-x));
}

// One wave32 per (b,h) chain. Lane i owns row i of the 32x32 state S.
__global__ __launch_bounds__(32)
void delta_cell_scan_kernel(const float* __restrict__ x,
                            const float* __restrict__ S0,
                            const float* __restrict__ scale_p,
                            const float* __restrict__ bias_p,
                            float* __restrict__ out) {
  __shared__ __align__(16) float tiles[NSTAGES * TILE_P];
  __shared__ __align__(16) float knbuf[N_];

  const int lane = threadIdx.x;            // 0..31 == row index i (and col index for v)
  const int bh   = blockIdx.x;             // 0..63
  const int b    = bh >> 3;
  const int h    = bh & 7;

  const float scale = scale_p[0];
  const float bias  = bias_p[0];

  // ---- load initial state row S[b,h,lane,:] into registers ----
  float Srow[N_];
  {
    const float4* sp =
        reinterpret_cast<const float4*>(S0 + ((size_t)(b * H_ + h) * N_ + lane) * N_);
#pragma unroll
    for (int q = 0; q < 8; ++q) {
      float4 v = sp[q];
      Srow[4 * q + 0] = v.x; Srow[4 * q + 1] = v.y;
      Srow[4 * q + 2] = v.z; Srow[4 * q + 3] = v.w;
    }
  }

  const float* xg = x + (size_t)b * D_ + (size_t)h * TILE_F;  // + t*xstep
  const size_t xstep = (size_t)B_ * D_;                       // 65536 floats / step

  AS3 float* tiles3 = (AS3 float*)tiles;

  // Issue one 4KB tile as 8 async B128 ops (32 lanes x 16B each), scattered
  // into the padded 36-float-stride LDS layout.
  auto issue_tile = [&](int t, int slot) {
    const float* g = xg + (size_t)t * xstep;
    AS3 float* l = tiles3 + slot * TILE_P;
#pragma unroll
    for (int q = 0; q < 8; ++q) {
      const int s   = q * 32 + lane;   // 16B segment id, 0..255
      const int row = s >> 3;          // 8 segments per 32-float row
      const int c4  = s & 7;
      async_copy_b128(g + s * 4, l + row * ROWP + c4 * 4);
    }
  };

#pragma unroll
  for (int st = 0; st < NSTAGES; ++st) issue_tile(st, st);

  for (int t = 0; t < T_; ++t) {
    // tile t done when at most (NSTAGES-1) tiles' worth of ops remain (in-order retire)
    WAIT_ASYNCCNT_LE((NSTAGES - 1) * 8);
    const int slot = t & (NSTAGES - 1);
    float* tile = tiles + slot * TILE_P;

    // ---- k_i: mean of row `lane` (8x b128 LDS reads, 2 accumulators) ----
    float rs0 = 0.f, rs1 = 0.f;
    {
      const float4* rp = reinterpret_cast<const float4*>(tile + lane * ROWP);
#pragma unroll
      for (int q = 0; q < 8; q += 2) {
        float4 a = rp[q], c = rp[q + 1];
        rs0 += (a.x + a.y) + (a.z + a.w);
        rs1 += (c.x + c.y) + (c.z + c.w);
      }
    }
    const float k_i = (rs0 + rs1) * (1.0f / 32.0f);

    // ---- v_lane: mean of column `lane` (conflict-free strided reads) ----
    float cs0 = 0.f, cs1 = 0.f, cs2 = 0.f, cs3 = 0.f;
#pragma unroll
    for (int i = 0; i < N_; i += 4) {
      cs0 += tile[(i + 0) * ROWP + lane];
      cs1 += tile[(i + 1) * ROWP + lane];
      cs2 += tile[(i + 2) * ROWP + lane];
      cs3 += tile[(i + 3) * ROWP + lane];
    }
    const float v_l = ((cs0 + cs1) + (cs2 + cs3)) * (1.0f / 32.0f);

    // ---- wave32 reductions: sum(k) and sum(k^2) ----
    float ksum = k_i, knorm2 = k_i * k_i;
#pragma unroll
    for (int m = 16; m >= 1; m >>= 1) {
      ksum   += __shfl_xor(ksum, m, 32);
      knorm2 += __shfl_xor(knorm2, m, 32);
    }
    const float A_mean = ksum * (1.0f / 32.0f);
    const float beta   = fast_sigmoid(scale * A_mean + bias);
    const float kn_i   = k_i / (sqrtf(knorm2) + 1e-6f);

    // ---- broadcast k_norm through LDS ----
    knbuf[lane] = kn_i;
    __syncthreads();
    float kn[N_];
    {
      const float4* kp = reinterpret_cast<const float4*>(knbuf);
#pragma unroll
      for (int q = 0; q < 8; ++q) {
        float4 v = kp[q];
        kn[4 * q + 0] = v.x; kn[4 * q + 1] = v.y;
        kn[4 * q + 2] = v.z; kn[4 * q + 3] = v.w;
      }
    }

    // ---- retrieved_i = <S_row, kn>  (4 accumulators) ----
    float r0 = 0.f, r1 = 0.f, r2 = 0.f, r3 = 0.f;
#pragma unroll
    for (int j = 0; j < N_; j += 4) {
      r0 = fmaf(Srow[j + 0], kn[j + 0], r0);
      r1 = fmaf(Srow[j + 1], kn[j + 1], r1);
      r2 = fmaf(Srow[j + 2], kn[j + 2], r2);
      r3 = fmaf(Srow[j + 3], kn[j + 3], r3);
    }
    const float delta = v_l - ((r0 + r1) + (r2 + r3));

    // All LDS reads of this slot are retired at this barrier; safe to
    // overwrite the slot with the next async tile.
    __syncthreads();
    if (t + NSTAGES < T_) issue_tile(t + NSTAGES, slot);

    // ---- S = tanh(beta*S + delta*kn);  Sq = <S_new, kn> ----
    float q0 = 0.f, q1 = 0.f;
#pragma unroll
    for (int j = 0; j < N_; j += 2) {
      float s0 = fast_tanh(fmaf(beta, Srow[j + 0], delta * kn[j + 0]));
      float s1 = fast_tanh(fmaf(beta, Srow[j + 1], delta * kn[j + 1]));
      Srow[j + 0] = s0;
      Srow[j + 1] = s1;
      q0 = fmaf(s0, kn[j + 0], q0);
      q1 = fmaf(s1, kn[j + 1], q1);
    }
    const float Sq = q0 + q1;

    // out = Sq * silu(Sq) = Sq^2 * sigmoid(Sq);  out[t, b, h*32 + lane]
    out[(size_t)t * (B_ * H_ * N_) + (size_t)b * (H_ * N_) + h * N_ + lane] =
        Sq * Sq * fast_sigmoid(Sq);
  }

  // ---- S_final[b,h,lane,:] after the scan output block ----
  float4* sf = reinterpret_cast<float4*>(
      out + (size_t)T_ * (B_ * H_ * N_) + ((size_t)(b * H_ + h) * N_ + lane) * N_);
#pragma unroll
  for (int q = 0; q < 8; ++q) {
    float4 v;
    v.x = Srow[4 * q + 0]; v.y = Srow[4 * q + 1];
    v.z = Srow[4 * q + 2]; v.w = Srow[4 * q + 3];
    sf[q] = v;
  }
}

extern "C" void kernel_launch(void* const* d_in, const int* in_sizes, int n_in,
                              void* d_out, int out_size, void* d_ws, size_t ws_size,
                              hipStream_t stream) {
  (void)in_sizes; (void)n_in; (void)d_ws; (void)ws_size; (void)out_size;
  const float* x     = (const float*)d_in[0];
  const float* S0    = (const float*)d_in[1];
  const float* scale = (const float*)d_in[2];
  const float* bias  = (const float*)d_in[3];
  float* out = (float*)d_out;

  dim3 grid(B_ * H_);   // 64 independent (b,h) scan chains
  dim3 block(32);       // one wave32 per chain
  hipLaunchKernelGGL(delta_cell_scan_kernel, grid, block, 0, stream,
                     x, S0, scale, bias, out);
}